// OuterProductNN_k3_25108378812917
// MI455X (gfx1250) — compile-verified
//
#include <hip/hip_runtime.h>
#include <hip/hip_bf16.h>
#include <math.h>

// Problem constants
#define NC   5      // N_COORDS
#define MM   35     // number of monomials  C(7,3)
#define XD   70     // feature dim = 2*MM
#define PD   80     // padded dim (5 tiles of 16, K padded to 80)
#define XSTR 81     // LDS row stride for X (81*4B -> conflict-free strided reads)
#define WAVES 4
#define ROWS_PER_BLOCK (WAVES * 16)

typedef float v2f __attribute__((ext_vector_type(2)));
typedef float v8f __attribute__((ext_vector_type(8)));

// combinations_with_replacement(range(5), 3), lexicographic
__constant__ int IDX_I[MM] = {0,0,0,0,0,0,0,0,0,0,0,0,0,0,0,
                              1,1,1,1,1,1,1,1,1,1,
                              2,2,2,2,2,2,
                              3,3,3,
                              4};
__constant__ int IDX_J[MM] = {0,0,0,0,0,1,1,1,1,2,2,2,3,3,4,
                              1,1,1,1,2,2,2,3,3,4,
                              2,2,2,3,3,4,
                              3,3,4,
                              4};
__constant__ int IDX_K[MM] = {0,1,2,3,4,1,2,3,4,2,3,4,3,4,4,
                              1,2,3,4,2,3,4,3,4,4,
                              2,3,4,3,4,4,
                              3,4,4,
                              4};

// Build zero-padded 80x80 Q from w (w[0:1225]=W_re row-major, w[1225:2450]=W_im).
// Q = [[ W_re, -W_im ],
//      [ W_im,  W_re ]]   so that  out = x^T Q x  with x=[mr;mi].
__global__ void build_q_kernel(const float* __restrict__ w, float* __restrict__ Q) {
    int idx = blockIdx.x * blockDim.x + threadIdx.x;
    if (idx >= PD * PD) return;
    int row = idx / PD;
    int col = idx % PD;
    float v = 0.0f;
    if (row < XD && col < XD) {
        if (row < MM && col < MM)      v =  w[row * MM + col];
        else if (row < MM)             v = -w[MM * MM + row * MM + (col - MM)];
        else if (col < MM)             v =  w[MM * MM + (row - MM) * MM + col];
        else                           v =  w[(row - MM) * MM + (col - MM)];
    }
    Q[idx] = v;
}

// One wave per 16-row batch tile: build X (monomials) in LDS, run
// T = X @ Q with V_WMMA_F32_16X16X4_F32, fuse rowwise dot(T, X), log, store.
__launch_bounds__(WAVES * 32)
__global__ void bilinear_wmma_kernel(const float* __restrict__ zr_g,
                                     const float* __restrict__ zi_g,
                                     const float* __restrict__ Qg,
                                     float* __restrict__ out, int B) {
    __shared__ float Qs[PD * PD];          // 25600 B
    __shared__ float Xs[WAVES][16][XSTR];  // 20736 B

    int tid = threadIdx.x;

    // Cooperative Q load (resident in 192MB L2 across blocks)
    for (int i = tid; i < PD * PD; i += blockDim.x) Qs[i] = Qg[i];

    int wave = tid >> 5;
    int lane = tid & 31;
    int r    = lane & 15;   // row within 16-row tile
    int half = lane >> 4;   // wave half (WMMA layout)
    int rowBase = blockIdx.x * ROWS_PER_BLOCK + wave * 16;
    int g = rowBase + r;

    // Load this row's coordinates
    float zr[NC], zi[NC];
    if (g < B) {
#pragma unroll
        for (int c = 0; c < NC; ++c) {
            zr[c] = zr_g[g * NC + c];
            zi[c] = zi_g[g * NC + c];
        }
    } else {
#pragma unroll
        for (int c = 0; c < NC; ++c) { zr[c] = 0.0f; zi[c] = 0.0f; }
    }

    // Monomials: half 0 computes t in [0,18), half 1 computes t in [18,35)
    int t0 = half ? 18 : 0;
    int t1 = half ? 35 : 18;
    for (int t = t0; t < t1; ++t) {
        int i = IDX_I[t], j = IDX_J[t], k = IDX_K[t];
        float pr = zr[i] * zr[j] - zi[i] * zi[j];
        float pi = zr[i] * zi[j] + zi[i] * zr[j];
        float mr = pr * zr[k] - pi * zi[k];
        float mi = pr * zi[k] + pi * zr[k];
        Xs[wave][r][t]      = mr;
        Xs[wave][r][MM + t] = mi;
    }
    // Zero padding columns 70..79 (half 0 -> 70..74, half 1 -> 75..79)
#pragma unroll
    for (int c = 0; c < 5; ++c) Xs[wave][r][XD + half * 5 + c] = 0.0f;

    __syncthreads();

    float acc[8];
#pragma unroll
    for (int v = 0; v < 8; ++v) acc[v] = 0.0f;

    // 5 column tiles x 20 K-steps of 16x16x4 fp32 WMMA
    for (int nt = 0; nt < 5; ++nt) {
        int n0 = nt * 16;
        v8f c = {0.f, 0.f, 0.f, 0.f, 0.f, 0.f, 0.f, 0.f};
#pragma unroll
        for (int ks = 0; ks < PD / 4; ++ks) {
            int k0 = ks * 4 + half * 2;
            v2f a, b;
            // A (16x4): lanes 0-15 hold K = {4ks, 4ks+1}, lanes 16-31 hold K = {4ks+2, 4ks+3}
            a.x = Xs[wave][r][k0];
            a.y = Xs[wave][r][k0 + 1];
            // B (4x16): mirrored striping, N = lane%16
            b.x = Qs[(k0)     * PD + n0 + r];
            b.y = Qs[(k0 + 1) * PD + n0 + r];
            c = __builtin_amdgcn_wmma_f32_16x16x4_f32(
                    /*neg_a=*/false, a, /*neg_b=*/false, b,
                    /*c_mod=*/(short)0, c, /*reuse_a=*/false, /*reuse_b=*/false);
        }
        // D layout: VGPR v -> row v + 8*half, column = n0 + (lane%16).
        // Fuse the rowwise dot T . X
#pragma unroll
        for (int v = 0; v < 8; ++v) {
            int m = v + half * 8;
            acc[v] += c[v] * Xs[wave][m][n0 + r];
        }
    }

    // Reduce the per-column partials across the 16 lanes of each half (wave32)
#pragma unroll
    for (int mask = 1; mask <= 8; mask <<= 1) {
#pragma unroll
        for (int v = 0; v < 8; ++v) acc[v] += __shfl_xor(acc[v], mask, 32);
    }

    if (r == 0) {
#pragma unroll
        for (int v = 0; v < 8; ++v) {
            int row = rowBase + v + half * 8;
            if (row < B) out[row] = logf(acc[v]);
        }
    }
}

extern "C" void kernel_launch(void* const* d_in, const int* in_sizes, int n_in,
                              void* d_out, int out_size, void* d_ws, size_t ws_size,
                              hipStream_t stream) {
    const float* zr = (const float*)d_in[0];   // (B, 5)
    const float* zi = (const float*)d_in[1];   // (B, 5)
    const float* w  = (const float*)d_in[2];   // (2450, 1)
    float* out = (float*)d_out;                // (B, 1)
    float* Q   = (float*)d_ws;                 // 80*80 floats = 25.6 KB scratch

    int B = in_sizes[0] / NC;

    build_q_kernel<<<(PD * PD + 255) / 256, 256, 0, stream>>>(w, Q);

    int blocks = (B + ROWS_PER_BLOCK - 1) / ROWS_PER_BLOCK;
    bilinear_wmma_kernel<<<blocks, WAVES * 32, 0, stream>>>(zr, zi, Q, out, B);
}